// CNF_9663676416569
// MI455X (gfx1250) — compile-verified
//
#include <hip/hip_runtime.h>
#include <cstddef>

// ---------------------------------------------------------------------------
// CNF vector field + exact divergence, fused for MI455X (gfx1250, wave32).
//
// div_i = u_i^T (W2 .* M) v_i   with M[p,q] = sum_j W1[j+1,p] W3[q,j],
// u = 1-h1^2, v = 1-h2^2.  K := W2 .* M is precomputed once (batch-free).
// Per 32-row batch tile:  A1 = X@W1x (+ t*W1[0]+b1) -> h1
//                         A2 = h1@W2 (+b2)          -> h2
//                         Y  = h2@W3 (+b3)
//                         Z  = (1-h1^2)@K ; div = rowsum(Z .* (1-h2^2))
// GEMMs: V_WMMA_F32_16X16X4_F32 (f32 matches reference precision).
// Stage C streams W2 panels via GLOBAL_LOAD_ASYNC_TO_LDS_B128 (ASYNCcnt);
// Stage E streams K panels via the Tensor Data Mover (TENSORcnt).
// Both are double-buffered so DMA overlaps WMMA.
// ---------------------------------------------------------------------------

typedef __attribute__((ext_vector_type(2))) float v2f;
typedef __attribute__((ext_vector_type(8))) float v8f;
typedef __attribute__((ext_vector_type(4))) unsigned tdm_g0_t;  // D# group0
typedef __attribute__((ext_vector_type(8))) unsigned tdm_g1_t;  // D# group1

#define CNF_H  512
#define CNF_D  64
#define CNF_MT 32      // batch rows per workgroup
#define CNF_B  8192

#if __has_builtin(__builtin_amdgcn_tanhf)
__device__ __forceinline__ float cnf_tanh(float x) { return __builtin_amdgcn_tanhf(x); }
#else
__device__ __forceinline__ float cnf_tanh(float x) { return tanhf(x); }
#endif

#if __has_builtin(__builtin_amdgcn_s_wait_asynccnt)
#define WAIT_ASYNC(n) __builtin_amdgcn_s_wait_asynccnt(n)
#else
#define WAIT_ASYNC(n) asm volatile("s_wait_asynccnt %0" ::"i"(n) : "memory")
#endif

__device__ __forceinline__ v8f wmma4(v2f a, v2f b, v8f c) {
  // D = A(16x4,f32) x B(4x16,f32) + C(16x16,f32)
  return __builtin_amdgcn_wmma_f32_16x16x4_f32(false, a, false, b, (short)0, c,
                                               false, false);
}

// Issue one 16x512-f32 K-panel (32 KB) as async Global->LDS DMA.
// 256 threads x 8 chunks; each instruction moves 32 lanes x 16B.
__device__ __forceinline__ void issue_panel_async(float* dst,
                                                  const float* __restrict__ src,
                                                  int tid) {
  const unsigned l0 = (unsigned)(size_t)dst;  // low 32 bits = LDS byte offset
#pragma unroll
  for (int j = 0; j < 8; ++j) {
    const unsigned off = (unsigned)((tid + j * 256) * 16);
    asm volatile("global_load_async_to_lds_b128 %0, %1, %2"
                 :
                 : "v"(l0 + off), "v"(off), "s"(src)
                 : "memory");
  }
}

// Issue one 16x512-f32 panel as a single Tensor Data Mover tile
// (2D tensor, 2 descriptor groups; issued by one wave, EXEC-independent).
__device__ __forceinline__ void tdm_load_panel(float* dstLds,
                                               const float* __restrict__ src) {
  const unsigned lds_addr = (unsigned)(size_t)dstLds;
  const unsigned long long ga = (unsigned long long)(size_t)src;
  tdm_g0_t g0;
  g0[0] = 1u;                                   // count=1 (valid user D#)
  g0[1] = lds_addr;                             // lds_addr [63:32]
  g0[2] = (unsigned)(ga & 0xffffffffu);         // global_addr [95:64]
  g0[3] = (unsigned)((ga >> 32) & 0x01ffffffu)  // global_addr [120:96]
          | (2u << 30);                         // type=2 ("image") [127:126]
  tdm_g1_t g1;
  g1[0] = (2u << 16);    // workgroup_mask=0 | data_size=2 (4 bytes)
  g1[1] = (512u << 16);  // atomic_barrier_addr=0 | tensor_dim0[15:0]=512
  g1[2] = (512u << 16);  // tensor_dim0[31:16]=0  | tensor_dim1[15:0]=512
  g1[3] = (512u << 16);  // tensor_dim1[31:16]=0  | tile_dim0=512
  g1[4] = 16u;           // tile_dim1=16 | tile_dim2=0
  g1[5] = 512u;          // tensor_dim0_stride[31:0]=512
  g1[6] = 0u;            // stride hi | tensor_dim1_stride lo
  g1[7] = 0u;            // tensor_dim1_stride hi
  asm volatile("tensor_load_to_lds %0, %1" ::"s"(g0), "s"(g1) : "memory");
}

// ---------------------------------------------------------------------------
// Kernel 0: K[p][q] = W2[p][q] * sum_j W1[(j+1)][p] * W3[q][j]   (512x512)
// ---------------------------------------------------------------------------
__global__ __launch_bounds__(256) void cnf_precompute_K(
    const float* __restrict__ W1, const float* __restrict__ W2,
    const float* __restrict__ W3, float* __restrict__ Kws) {
  const int idx = blockIdx.x * 256 + threadIdx.x;  // 0 .. 512*512-1
  const int p = idx >> 9;
  const int q = idx & 511;
  float s = 0.f;
#pragma unroll 8
  for (int j = 0; j < CNF_D; ++j)
    s += W1[(size_t)(j + 1) * CNF_H + p] * W3[(size_t)q * CNF_D + j];
  Kws[idx] = s * W2[idx];
}

// ---------------------------------------------------------------------------
// Kernel 1: fused forward + divergence for a tile of 32 batch rows.
// 256 threads = 8 waves. Wave w owns N-columns [w*64, w*64+64) as 2x4 tiles.
// ---------------------------------------------------------------------------
__global__ __launch_bounds__(256) void cnf_fused(
    const float* __restrict__ t_p, const float* __restrict__ x,
    const float* __restrict__ W1, const float* __restrict__ b1,
    const float* __restrict__ W2, const float* __restrict__ b2,
    const float* __restrict__ W3, const float* __restrict__ b3,
    const float* __restrict__ Kmat, float* __restrict__ out) {
  extern __shared__ float lds[];
  float* H1     = lds;                        // 32 x 512
  float* H2     = H1 + CNF_MT * CNF_H;        // 32 x 512
  float* panelA = H2 + CNF_MT * CNF_H;        // 16 x 512 (K-panel buf 0)
  float* panelB = panelA + 16 * CNF_H;        // 16 x 512 (K-panel buf 1)
  float* Xs     = panelB + 16 * CNF_H;        // 32 x 64
  float* divAcc = Xs + CNF_MT * CNF_D;        // 32

  const int tid  = threadIdx.x;
  const int lane = tid & 31;
  const int w    = tid >> 5;   // wave id 0..7
  const int hl   = lane >> 4;  // K-half select (0/1)
  const int lm   = lane & 15;  // M (A) / N (B,C,D) index
  const int blockBase = blockIdx.x * CNF_MT;
  const float tval = t_p[0];

  // Prefetch first W2 K-panel; DMA overlaps X staging + Stage A.
  issue_panel_async(panelA, W2, tid);

  if (tid < CNF_MT) divAcc[tid] = 0.f;

  // Stage X tile (rows stride D+1=65 in global -> dense 32x64 in LDS)
  for (int idx = tid; idx < CNF_MT * CNF_D; idx += 256) {
    const int m = idx >> 6, k = idx & 63;
    Xs[idx] = x[(size_t)(blockBase + m) * (CNF_D + 1) + k];
  }
  __syncthreads();

  // ---------------- Stage A: H1 = tanh(X @ W1x + t*W1[0] + b1) -------------
  {
    v8f acc[2][4] = {};
    for (int k0 = 0; k0 < CNF_D; k0 += 4) {
      v2f a[2];
#pragma unroll
      for (int r = 0; r < 2; ++r) {
        const float* ap = &Xs[(r * 16 + lm) * CNF_D + k0 + 2 * hl];
        a[r].x = ap[0];
        a[r].y = ap[1];
      }
#pragma unroll
      for (int c = 0; c < 4; ++c) {
        const int n = (w * 4 + c) * 16 + lm;
        v2f b;
        b.x = W1[(size_t)(k0 + 2 * hl + 1) * CNF_H + n];  // W1x row k -> W1 row k+1
        b.y = W1[(size_t)(k0 + 2 * hl + 2) * CNF_H + n];
#pragma unroll
        for (int r = 0; r < 2; ++r) acc[r][c] = wmma4(a[r], b, acc[r][c]);
      }
    }
#pragma unroll
    for (int c = 0; c < 4; ++c) {
      const int n = (w * 4 + c) * 16 + lm;
      const float cadd = tval * W1[n] + b1[n];  // t*W1[0] + bias
#pragma unroll
      for (int r = 0; r < 2; ++r)
#pragma unroll
        for (int i = 0; i < 8; ++i) {
          const int row = r * 16 + i + 8 * hl;
          H1[row * CNF_H + n] = cnf_tanh(acc[r][c][i] + cadd);
        }
    }
  }

  // ---------------- Stage C: H2 = tanh(H1 @ W2 + b2) -----------------------
  // Double-buffered async K-panels: issue next panel, wait current, compute.
  {
    v8f acc[2][4] = {};
    for (int kb = 0; kb < CNF_H; kb += 16) {
      float* cur = (kb & 16) ? panelB : panelA;
      if (kb + 16 < CNF_H) {
        issue_panel_async((kb & 16) ? panelA : panelB,
                          W2 + (size_t)(kb + 16) * CNF_H, tid);
        WAIT_ASYNC(8);  // loads retire in order -> current panel complete
      } else {
        WAIT_ASYNC(0);
      }
      __syncthreads();  // all waves' panel chunks visible (fences H1 too)
#pragma unroll
      for (int ks = 0; ks < 4; ++ks) {
        const int k0 = ks * 4;
        v2f a[2];
#pragma unroll
        for (int r = 0; r < 2; ++r) {
          const float* ap = &H1[(r * 16 + lm) * CNF_H + kb + k0 + 2 * hl];
          a[r].x = ap[0];
          a[r].y = ap[1];
        }
#pragma unroll
        for (int c = 0; c < 4; ++c) {
          const int n = (w * 4 + c) * 16 + lm;
          v2f b;
          b.x = cur[(k0 + 2 * hl) * CNF_H + n];
          b.y = cur[(k0 + 2 * hl + 1) * CNF_H + n];
#pragma unroll
          for (int r = 0; r < 2; ++r) acc[r][c] = wmma4(a[r], b, acc[r][c]);
        }
      }
      __syncthreads();  // all waves done reading cur before its next overwrite
    }
#pragma unroll
    for (int c = 0; c < 4; ++c) {
      const int n = (w * 4 + c) * 16 + lm;
      const float bb = b2[n];
#pragma unroll
      for (int r = 0; r < 2; ++r)
#pragma unroll
        for (int i = 0; i < 8; ++i) {
          const int row = r * 16 + i + 8 * hl;
          H2[row * CNF_H + n] = cnf_tanh(acc[r][c][i] + bb);
        }
    }
  }
  __syncthreads();  // H2 visible to all waves

  // Prefetch first Kmat panel via the Tensor Data Mover (one wave issues a
  // single 32 KB 2D-tile DMA); it flies during all of Stage D.
  if (w == 0) tdm_load_panel(panelA, Kmat);

  // ---------------- Stage D: Y = H2 @ W3 + b3  (one 16x16 tile per wave) ---
  {
    const int r = w >> 2, c = w & 3;
    const int n = c * 16 + lm;
    v8f acc = {};
#pragma unroll 8
    for (int k0 = 0; k0 < CNF_H; k0 += 4) {
      v2f a, b;
      const float* ap = &H2[(r * 16 + lm) * CNF_H + k0 + 2 * hl];
      a.x = ap[0];
      a.y = ap[1];
      b.x = W3[(size_t)(k0 + 2 * hl) * CNF_D + n];
      b.y = W3[(size_t)(k0 + 2 * hl + 1) * CNF_D + n];
      acc = wmma4(a, b, acc);
    }
    const float bb = b3[n];
#pragma unroll
    for (int i = 0; i < 8; ++i) {
      const size_t gRow = (size_t)(blockBase + r * 16 + i + 8 * hl);
      out[gRow * (CNF_D + 1) + n] = acc[i] + bb;
    }
  }

  // ---------------- Stage E: Z = (1-H1^2) @ K ; div = rowsum(Z .* (1-H2^2))
  // TDM double-buffering: wave 0 issues next tile, waits TENSORcnt<=1
  // (per-wave TDM ops complete in order -> current panel resident), barrier.
  {
    v8f acc[2][4] = {};
    for (int kb = 0; kb < CNF_H; kb += 16) {
      float* cur = (kb & 16) ? panelB : panelA;
      if (w == 0) {
        if (kb + 16 < CNF_H) {
          tdm_load_panel((kb & 16) ? panelA : panelB,
                         Kmat + (size_t)(kb + 16) * CNF_H);
          __builtin_amdgcn_s_wait_tensorcnt((short)1);
        } else {
          __builtin_amdgcn_s_wait_tensorcnt((short)0);
        }
      }
      __syncthreads();  // publish current panel to all waves
#pragma unroll
      for (int ks = 0; ks < 4; ++ks) {
        const int k0 = ks * 4;
        v2f a[2];
#pragma unroll
        for (int r = 0; r < 2; ++r) {
          const float* ap = &H1[(r * 16 + lm) * CNF_H + kb + k0 + 2 * hl];
          const float h0 = ap[0], h1v = ap[1];
          a[r].x = 1.f - h0 * h0;  // u = 1 - h1^2
          a[r].y = 1.f - h1v * h1v;
        }
#pragma unroll
        for (int c = 0; c < 4; ++c) {
          const int n = (w * 4 + c) * 16 + lm;
          v2f b;
          b.x = cur[(k0 + 2 * hl) * CNF_H + n];
          b.y = cur[(k0 + 2 * hl + 1) * CNF_H + n];
#pragma unroll
          for (int r = 0; r < 2; ++r) acc[r][c] = wmma4(a[r], b, acc[r][c]);
        }
      }
      __syncthreads();  // all waves done reading cur before its next overwrite
    }
    // Z .* V, reduce columns: per-lane partials -> shuffle tree over 16 lanes
#pragma unroll
    for (int r = 0; r < 2; ++r)
#pragma unroll
      for (int i = 0; i < 8; ++i) {
        const int row = r * 16 + i + 8 * hl;
        float s = 0.f;
#pragma unroll
        for (int c = 0; c < 4; ++c) {
          const int n = (w * 4 + c) * 16 + lm;
          const float h2v = H2[row * CNF_H + n];
          s += acc[r][c][i] * (1.f - h2v * h2v);
        }
        s += __shfl_xor(s, 1, 32);
        s += __shfl_xor(s, 2, 32);
        s += __shfl_xor(s, 4, 32);
        s += __shfl_xor(s, 8, 32);
        if (lm == 0) atomicAdd(&divAcc[row], s);
      }
  }
  __syncthreads();
  if (tid < CNF_MT)
    out[(size_t)(blockBase + tid) * (CNF_D + 1) + CNF_D] = divAcc[tid];
}

// ---------------------------------------------------------------------------
extern "C" void kernel_launch(void* const* d_in, const int* in_sizes, int n_in,
                              void* d_out, int out_size, void* d_ws,
                              size_t ws_size, hipStream_t stream) {
  const float* t  = (const float*)d_in[0];
  const float* x  = (const float*)d_in[1];
  const float* W1 = (const float*)d_in[2];
  const float* b1 = (const float*)d_in[3];
  const float* W2 = (const float*)d_in[4];
  const float* b2 = (const float*)d_in[5];
  const float* W3 = (const float*)d_in[6];
  const float* b3 = (const float*)d_in[7];
  float* out = (float*)d_out;
  float* Kws = (float*)d_ws;  // 512*512 floats = 1 MB scratch

  cnf_precompute_K<<<(CNF_H * CNF_H) / 256, 256, 0, stream>>>(W1, W2, W3, Kws);

  const size_t smem =
      (size_t)(2 * CNF_MT * CNF_H + 2 * 16 * CNF_H + CNF_MT * CNF_D + CNF_MT) *
      sizeof(float);
  cnf_fused<<<CNF_B / CNF_MT, 256, smem, stream>>>(t, x, W1, b1, W2, b2, W3,
                                                   b3, Kws, out);
}